// HardPartPyramidPooling_27324581937298
// MI455X (gfx1250) — compile-verified
//
#include <hip/hip_runtime.h>
#include <hip/hip_bf16.h>

typedef __attribute__((ext_vector_type(2))) float v2f;
typedef __attribute__((ext_vector_type(4))) float v4f;
typedef __attribute__((ext_vector_type(8))) float v8f;
typedef __attribute__((ext_vector_type(4))) int   v4i;

#define HPP_NEG_FILL (-100.0f)

namespace {
constexpr int kN  = 16;
constexpr int kC  = 256;
constexpr int kS  = 32;
constexpr int kH  = 16;
constexpr int kW  = 11;
constexpr int kHW = kH * kW;     // 176
constexpr int kP  = 16;          // parts
constexpr int kNS = kN * kS;     // 512 samples
constexpr int kCT = kC / 16;     // 16 channel tiles of 16
}

// LDS float max via native ds_max_num_f32 path
__device__ __forceinline__ void lds_fmax(float* p, float v) {
  __hip_atomic_fetch_max(p, v, __ATOMIC_RELAXED, __HIP_MEMORY_SCOPE_WORKGROUP);
}

// One wave (32 lanes) per block. Block b handles sample ns = b>>4 and
// channel tile ct = b&15 (16 channels x 176 positions -> 16x16 outputs).
__global__ __launch_bounds__(32)
void hppp_kernel(const float* __restrict__ x,
                 const int* __restrict__ labels,
                 float* __restrict__ out) {
  __shared__ alignas(16) int   sh_lbl[kHW];  // labels for this sample
  __shared__ float sh_max[16 * kP];          // per-(channel, part) running max
  __shared__ float sh_cnt[kP];               // per-part counts
#if !__has_builtin(__builtin_amdgcn_wmma_f32_16x16x4_f32)
  __shared__ float sh_sum[16 * kP];          // fallback sum accumulator
#endif

  const int lane = threadIdx.x;        // 0..31
  const int blk  = blockIdx.x;
  const int ns   = blk >> 4;           // 0..511
  const int ct   = blk & 15;           // channel tile
  const int n_i  = ns >> 5;            // / kS
  const int s_i  = ns & 31;            // % kS

  // ---- init LDS ----
  if (lane < kP) sh_cnt[lane] = 0.0f;
  #pragma unroll
  for (int i = lane; i < 16 * kP; i += 32) {
    sh_max[i] = HPP_NEG_FILL;
#if !__has_builtin(__builtin_amdgcn_wmma_f32_16x16x4_f32)
    sh_sum[i] = 0.0f;
#endif
  }

  // ---- labels (shared across the 16 channels of this tile) ----
  const int* __restrict__ lbl = labels + (size_t)(n_i * kS + s_i) * kHW;
  #pragma unroll
  for (int i = lane; i < kHW; i += 32) sh_lbl[i] = lbl[i];
  __syncthreads();

  // ---- counts ----
  #pragma unroll
  for (int i = lane; i < kHW; i += 32) atomicAdd(&sh_cnt[sh_lbl[i]], 1.0f);

  // ---- main pass: b128 streaming loads + WMMA sum + LDS atomic max ----
  // Each lane owns channel m = lane&15.  Per 8-position group, lanes 0-15
  // load positions k0..k0+3 and lanes 16-31 load k0+4..k0+7 (one b128 each).
  // WMMA contracts A's (lane,slot) K-index against B's identical mapping, so
  // any consistent position->K bijection yields the correct segmented sum:
  //   WMMA #1 covers {k0,k0+1 | k0+4,k0+5}, WMMA #2 covers the other four.
  const int m     = lane & 15;         // channel-in-tile (A row) / part col (B)
  const int half  = lane >> 4;
  const int koff4 = half * 4;
  const float* __restrict__ row =
      x + ((size_t)(n_i * kC + (ct * 16 + m)) * kS + s_i) * kHW;
  const v4f* __restrict__ row4 = reinterpret_cast<const v4f*>(row);

  v8f acc = {};
  #pragma unroll 2
  for (int k0 = 0; k0 < kHW; k0 += 8) {           // 22 iterations
    const int p0 = k0 + koff4;                    // 16B aligned by construction
    v4f a = __builtin_nontemporal_load(&row4[p0 >> 2]);   // streaming 16B read
    v4i l = *reinterpret_cast<const v4i*>(&sh_lbl[p0]);   // ds_load_b128

    // segmented max (each lane owns channel m of the tile)
    lds_fmax(&sh_max[m * kP + l.x], a.x);
    lds_fmax(&sh_max[m * kP + l.y], a.y);
    lds_fmax(&sh_max[m * kP + l.z], a.z);
    lds_fmax(&sh_max[m * kP + l.w], a.w);

#if __has_builtin(__builtin_amdgcn_wmma_f32_16x16x4_f32)
    v2f A0 = {a.x, a.y};
    v2f B0 = {(l.x == m) ? 1.0f : 0.0f, (l.y == m) ? 1.0f : 0.0f};
    acc = __builtin_amdgcn_wmma_f32_16x16x4_f32(
        false, A0, false, B0, (short)0, acc, false, false);
    v2f A1 = {a.z, a.w};
    v2f B1 = {(l.z == m) ? 1.0f : 0.0f, (l.w == m) ? 1.0f : 0.0f};
    acc = __builtin_amdgcn_wmma_f32_16x16x4_f32(
        false, A1, false, B1, (short)0, acc, false, false);
#else
    atomicAdd(&sh_sum[m * kP + l.x], a.x);
    atomicAdd(&sh_sum[m * kP + l.y], a.y);
    atomicAdd(&sh_sum[m * kP + l.z], a.z);
    atomicAdd(&sh_sum[m * kP + l.w], a.w);
#endif
  }
  __syncthreads();

  // ---- epilogue: mean + masked max, store transposed (n, c, s, p) ----
  // C/D 16x16 f32 layout: VGPR r -> M = r (lanes 0-15) or r+8 (lanes 16-31),
  // N = lane&15.  So this lane's part index is p = m.
  const float cnt   = sh_cnt[m];
  const float denom = fmaxf(cnt, 1.0f);
  #pragma unroll
  for (int r = 0; r < 8; ++r) {
    const int cit = r + half * 8;                  // channel-in-tile
#if __has_builtin(__builtin_amdgcn_wmma_f32_16x16x4_f32)
    const float sum = acc[r];
#else
    const float sum = sh_sum[cit * kP + m];
#endif
    float mx  = sh_max[cit * kP + m];
    mx        = (cnt > 0.0f) ? mx : 0.0f;
    const float val = sum / denom + mx;
    out[((size_t)(n_i * kC + (ct * 16 + cit)) * kS + s_i) * kP + m] = val;
  }
}

extern "C" void kernel_launch(void* const* d_in, const int* in_sizes, int n_in,
                              void* d_out, int out_size, void* d_ws, size_t ws_size,
                              hipStream_t stream) {
  (void)in_sizes; (void)n_in; (void)out_size; (void)d_ws; (void)ws_size;
  const float* x      = (const float*)d_in[0];
  const int*   labels = (const int*)d_in[1];   // parts_num (d_in[2]) fixed at 16
  float*       out    = (float*)d_out;

  const int blocks = kNS * kCT;                 // 512 * 16 = 8192 waves
  hppp_kernel<<<blocks, 32, 0, stream>>>(x, labels, out);
}